// HybridLoss_85152021611310
// MI455X (gfx1250) — compile-verified
//
#include <hip/hip_runtime.h>

typedef float v2f __attribute__((ext_vector_type(2)));
typedef float v8f __attribute__((ext_vector_type(8)));

#define FRCP(x)  __builtin_amdgcn_rcpf(x)    // v_rcp_f32  (~1 ulp)
#define FRSQ(x)  __builtin_amdgcn_rsqf(x)    // v_rsq_f32  (~1 ulp)
#define FSQRT(x) __builtin_amdgcn_sqrtf(x)   // v_sqrt_f32 (~1 ulp)

struct Quat { float x, y, z, w; };

__device__ __forceinline__ Quat hamilton(Quat a, Quat b) {
    Quat r;
    r.w = a.w*b.w - a.x*b.x - a.y*b.y - a.z*b.z;
    r.x = a.w*b.x + a.x*b.w + a.y*b.z - a.z*b.y;
    r.y = a.w*b.y - a.x*b.z + a.y*b.w + a.z*b.x;
    r.z = a.w*b.z + a.x*b.y - a.y*b.x + a.z*b.w;
    return r;
}

// _canonicalize: normalize quat (+1e-6), sign-fix by w, stable argsort(-dims),
// conjugate by permutation quaternion, flip xyz, renormalize.
__device__ __forceinline__ void canonicalize(const float* p, float* c, float* d, Quat& qout) {
    c[0] = p[0]; c[1] = p[1]; c[2] = p[2];
    float dx = p[3], dy = p[4], dz = p[5];
    Quat q; q.x = p[6]; q.y = p[7]; q.z = p[8]; q.w = p[9];
    float n = FSQRT(q.x*q.x + q.y*q.y + q.z*q.z + q.w*q.w) + 1e-6f;
    float inv = FRCP(n);
    q.x *= inv; q.y *= inv; q.z *= inv; q.w *= inv;
    float s = (q.w > 0.0f) ? 1.0f : ((q.w < 0.0f) ? -1.0f : 0.0f);
    q.x *= s; q.y *= s; q.z *= s; q.w *= s;

    const float S = 0.7071f;
    Quat pq;
    // stable descending sort of dims; perm -> PERM_TABLE entry (xyzw)
    if (dx >= dy) {
        if (dy >= dz)      { d[0]=dx; d[1]=dy; d[2]=dz; pq = Quat{0.f, 0.f, 0.f, 1.f}; }  // (0,1,2)
        else if (dx >= dz) { d[0]=dx; d[1]=dz; d[2]=dy; pq = Quat{0.f,   S, 0.f,   S}; }  // (0,2,1)
        else               { d[0]=dz; d[1]=dx; d[2]=dy; pq = Quat{0.f,  -S, 0.f,   S}; }  // (2,0,1)
    } else {
        if (dx >= dz)      { d[0]=dy; d[1]=dx; d[2]=dz; pq = Quat{  S, 0.f, 0.f,   S}; }  // (1,0,2)
        else if (dy >= dz) { d[0]=dy; d[1]=dz; d[2]=dx; pq = Quat{0.f, 0.f,   S,   S}; }  // (1,2,0)
        else               { d[0]=dz; d[1]=dy; d[2]=dx; pq = Quat{0.f, 0.f,  -S,   S}; }  // (2,1,0)
    }
    Quat t = hamilton(pq, q);
    // reference builds perm_q_inv = stack([w,-x,-y,-z]) placed in xyzw slots (verbatim)
    Quat pqi; pqi.x = pq.w; pqi.y = -pq.x; pqi.z = -pq.y; pqi.w = -pq.z;
    Quat qc = hamilton(t, pqi);
    qc.x = -qc.x; qc.y = -qc.y; qc.z = -qc.z;   // * [-1,-1,-1,1]
    float i2 = FRSQ(qc.x*qc.x + qc.y*qc.y + qc.z*qc.z + qc.w*qc.w);
    qout.x = qc.x*i2; qout.y = qc.y*i2; qout.z = qc.z*i2; qout.w = qc.w*i2;
}

// Wave(32)-wide sum via V_WMMA_F32_16X16X4_F32 with data in A, ones in B.
// A layout (documented): VGPR0 lanes0-15 -> (M=lane,K=0), lanes16-31 -> (M=lane-16,K=2);
// VGPR1 = 0 -> K=1,3 rows are zero.  So D[m][n] = v_m + v_{m+16} for every n.
// D layout: lane L<16 : d[j] = v_j     + v_{j+16}   (j = 0..7)
//           lane L>=16: d[j] = v_{j+8} + v_{j+24}
// => in-lane sum of the 8 D regs gives each half's partial; one hardware
//    SWAPX16 ds_swizzle (+add) yields the full 32-lane total in every lane.
__device__ __forceinline__ float wave_sum_wmma(float v) {
    v2f a; a[0] = v;    a[1] = 0.0f;
    v2f b; b[0] = 1.0f; b[1] = 1.0f;
    v8f cz = {0.f,0.f,0.f,0.f,0.f,0.f,0.f,0.f};
    v8f dd = __builtin_amdgcn_wmma_f32_16x16x4_f32(false, a, false, b, (short)0, cz, false, false);
    float t = ((dd[0] + dd[1]) + (dd[2] + dd[3])) + ((dd[4] + dd[5]) + (dd[6] + dd[7]));
    // SWAPX16: offset[14:10]=xor=0x10, or=0, and=0x1f -> 0x401f
    float u = __int_as_float(__builtin_amdgcn_ds_swizzle(__float_as_int(t), 0x401f));
    return t + u;
}

__global__ __launch_bounds__(256) void hybrid_loss_main(
    const float* __restrict__ pred, const float* __restrict__ gt,
    float* __restrict__ partials, int B) {

    const int i = blockIdx.x * 256 + threadIdx.x;
    const bool valid = (i < B);
    const int idx = valid ? i : (B - 1);

    // 40-byte records, 8-byte aligned: 5 x b64 loads per input, fully coalesced.
    float pp[10], gg[10];
    const float2* pv = reinterpret_cast<const float2*>(pred) + (size_t)idx * 5;
    const float2* gv = reinterpret_cast<const float2*>(gt)   + (size_t)idx * 5;
#pragma unroll
    for (int k = 0; k < 5; ++k) {
        float2 a = pv[k]; pp[2*k] = a.x; pp[2*k+1] = a.y;
        float2 b = gv[k]; gg[2*k] = b.x; gg[2*k+1] = b.y;
    }

    float pc[3], pd[3], gc[3], gd[3];
    Quat pq, gq;
    canonicalize(pp, pc, pd, pq);
    canonicalize(gg, gc, gd, gq);

    // ---- weighted params loss terms ----
    float diag = FSQRT(gd[0]*gd[0] + gd[1]*gd[1] + gd[2]*gd[2]) + 1e-6f;
    float idiag = FRCP(diag);
    float c_term = (fabsf(pc[0]-gc[0]) + fabsf(pc[1]-gc[1]) + fabsf(pc[2]-gc[2])) * idiag;
    float ig0 = FRCP(gd[0]+1e-6f), ig1 = FRCP(gd[1]+1e-6f), ig2 = FRCP(gd[2]+1e-6f);
    float d_term = fabsf(pd[0]-gd[0]) * ig0
                 + fabsf(pd[1]-gd[1]) * ig1
                 + fabsf(pd[2]-gd[2]) * ig2;
    float pn = FSQRT(pq.x*pq.x + pq.y*pq.y + pq.z*pq.z + pq.w*pq.w) + 1e-6f;
    float gn = FSQRT(gq.x*gq.x + gq.y*gq.y + gq.z*gq.z + gq.w*gq.w) + 1e-6f;
    float dot = (pq.x*gq.x + pq.y*gq.y + pq.z*gq.z + pq.w*gq.w) * FRCP(pn * gn);
    dot = fminf(fmaxf(fabsf(dot), 1e-6f), 1.0f);
    float q_term = 2.0f * acosf(dot);

    // ---- relative quat (verbatim reference arithmetic) ----
    float rp = FRSQ(pq.x*pq.x + pq.y*pq.y + pq.z*pq.z + pq.w*pq.w);
    float rg = FRSQ(gq.x*gq.x + gq.y*gq.y + gq.z*gq.z + gq.w*gq.w);
    float pqx = pq.x*rp, pqy = pq.y*rp, pqz = pq.z*rp, pqw = pq.w*rp;
    float gqx = gq.x*rg, gqy = gq.y*rg, gqz = gq.z*rg, gqw = gq.w*rg;
    float a0 = gqw, a1 = -gqx, a2 = -gqy, a3 = -gqz;
    float b0 = pqx, b1 = pqy, b2 = pqz, b3 = pqw;
    float r0 = a0*b0 - a1*b1 - a2*b2 - a3*b3;
    float r1 = a0*b1 + a1*b0 + a2*b3 - a3*b2;
    float r2 = a0*b2 - a1*b3 + a2*b0 + a3*b1;
    float r3 = a0*b3 + a1*b2 - a2*b1 + a3*b0;
    float rn = FRSQ(r0*r0 + r1*r1 + r2*r2 + r3*r3);
    float qx = r0*rn, qy = r1*rn, qz = r2*rn, qw = r3*rn;  // xyzw

    // rotation matrix of rel quat
    float R00 = 1.f - 2.f*(qy*qy + qz*qz), R01 = 2.f*(qx*qy - qz*qw), R02 = 2.f*(qx*qz + qy*qw);
    float R10 = 2.f*(qx*qy + qz*qw), R11 = 1.f - 2.f*(qx*qx + qz*qz), R12 = 2.f*(qy*qz - qx*qw);
    float R20 = 2.f*(qx*qz - qy*qw), R21 = 2.f*(qy*qz + qx*qw), R22 = 1.f - 2.f*(qx*qx + qy*qy);

    // per-axis extents of the 8 corners (signs independent -> abs trick), then
    // scale by 1/(gt_dims+1e-6) (positive, preserves min/max)
    float cc0 = pc[0]-gc[0], cc1 = pc[1]-gc[1], cc2 = pc[2]-gc[2];
    float e0 = 0.5f*(fabsf(R00)*pd[0] + fabsf(R01)*pd[1] + fabsf(R02)*pd[2]);
    float e1 = 0.5f*(fabsf(R10)*pd[0] + fabsf(R11)*pd[1] + fabsf(R12)*pd[2]);
    float e2 = 0.5f*(fabsf(R20)*pd[0] + fabsf(R21)*pd[1] + fabsf(R22)*pd[2]);
    float mn[3], mx[3];
    mn[0] = (cc0-e0)*ig0; mx[0] = (cc0+e0)*ig0;
    mn[1] = (cc1-e1)*ig1; mx[1] = (cc1+e1)*ig1;
    mn[2] = (cc2-e2)*ig2; mx[2] = (cc2+e2)*ig2;

    // gt box is the unit cube [-0.5, 0.5]^3
    float il[3], l1[3], hv[3], hu[3];
#pragma unroll
    for (int a = 0; a < 3; ++a) {
        float lo = fmaxf(mn[a], -0.5f);
        float hi = fminf(mx[a],  0.5f);
        il[a] = fmaxf(hi - lo + 0.001f, 0.0f);
        l1[a] = mx[a] - mn[a] + 0.001f;
        hv[a] = fmaxf(hi - lo, 0.0f);
        hu[a] = (mx[a] - mn[a]) + 1.0f - hv[a];
    }
    const float AREA2 = 1.002001f;  // (1+0.001)^2
    float iou = 1e30f;
    const int U[3] = {0, 0, 1}, V[3] = {2, 1, 2}, W[3] = {1, 2, 0};
#pragma unroll
    for (int pI = 0; pI < 3; ++pI) {
        float inter = il[U[pI]] * il[V[pI]];
        float iou2  = inter * FRCP(l1[U[pI]] * l1[V[pI]] + AREA2 - inter + 1e-6f);
        float iouh  = hv[W[pI]] * FRCP(hu[W[pI]] + 1e-6f);
        iou = fminf(iou, iou2 * iouh);
    }
    iou = fminf(fmaxf(iou, 0.0f), 1.0f);

    // mask padded threads (no divergence: EXEC stays full for WMMA)
    float m = valid ? 1.0f : 0.0f;
    c_term *= m; d_term *= m; q_term *= m; iou *= m;

    // wave-wide sums on the matrix pipe, then cross-wave via LDS
    float s0 = wave_sum_wmma(c_term);
    float s1 = wave_sum_wmma(d_term);
    float s2 = wave_sum_wmma(q_term);
    float s3 = wave_sum_wmma(iou);

    __shared__ float smem[8][4];
    const int lane = threadIdx.x & 31;
    const int wid  = threadIdx.x >> 5;
    if (lane == 0) { smem[wid][0] = s0; smem[wid][1] = s1; smem[wid][2] = s2; smem[wid][3] = s3; }
    __syncthreads();
    if (threadIdx.x < 4) {
        float acc = 0.0f;
#pragma unroll
        for (int w = 0; w < 8; ++w) acc += smem[w][threadIdx.x];
        partials[(size_t)blockIdx.x * 4 + threadIdx.x] = acc;
    }
}

__global__ __launch_bounds__(256) void hybrid_loss_finish(
    const float* __restrict__ partials, int nparts, float* __restrict__ out, int B) {
    __shared__ double sm[256][4];
    double s0 = 0.0, s1 = 0.0, s2 = 0.0, s3 = 0.0;
    for (int k = threadIdx.x; k < nparts; k += 256) {
        s0 += (double)partials[4*k + 0];
        s1 += (double)partials[4*k + 1];
        s2 += (double)partials[4*k + 2];
        s3 += (double)partials[4*k + 3];
    }
    sm[threadIdx.x][0] = s0; sm[threadIdx.x][1] = s1;
    sm[threadIdx.x][2] = s2; sm[threadIdx.x][3] = s3;
    __syncthreads();
    for (int off = 128; off > 0; off >>= 1) {
        if ((int)threadIdx.x < off) {
#pragma unroll
            for (int q = 0; q < 4; ++q) sm[threadIdx.x][q] += sm[threadIdx.x + off][q];
        }
        __syncthreads();
    }
    if (threadIdx.x == 0) {
        double Bd = (double)B;
        double loss = 0.4 * sm[0][0] / (3.0 * Bd)
                    + 0.4 * sm[0][1] / (3.0 * Bd)
                    + 0.2 * sm[0][2] / Bd
                    + 0.5 * (1.0 - sm[0][3] / Bd);
        out[0] = (float)loss;
    }
}

extern "C" void kernel_launch(void* const* d_in, const int* in_sizes, int n_in,
                              void* d_out, int out_size, void* d_ws, size_t ws_size,
                              hipStream_t stream) {
    const float* pred = (const float*)d_in[0];
    const float* gt   = (const float*)d_in[1];
    const int B = in_sizes[0] / 10;               // (B,10) float32
    const int nblocks = (B + 255) / 256;          // 4096 for B = 2^20
    float* partials = (float*)d_ws;               // nblocks*4 floats (64 KB)

    hybrid_loss_main<<<nblocks, 256, 0, stream>>>(pred, gt, partials, B);
    hybrid_loss_finish<<<1, 256, 0, stream>>>(partials, nblocks, (float*)d_out, B);
}